// RewardGatingModel_26603027431772
// MI455X (gfx1250) — compile-verified
//
#include <hip/hip_runtime.h>

// Problem constants (match reference)
#define BATCH 1024
#define TT    1024
#define DOBS  32
#define DACT  4
#define DIN   36     // DOBS + DACT
#define HH    64
#define GG    192    // 3*H

typedef float v2f __attribute__((ext_vector_type(2)));
typedef float v8f __attribute__((ext_vector_type(8)));

// LDS strides chosen to avoid bank conflicts for the split-half (k / k+2)
// WMMA A-operand reads and the D-tile writes.
#define XS 37    // x tile stride (odd)
#define HS 67    // h tile stride (odd)
#define SS 138   // r/z sum tile stride (8*138 % 64 == 16 -> disjoint halves on D write)
#define NS 66    // gi_n / gh_n tile stride (8*66 % 64 == 16)

__launch_bounds__(384)
__global__ void gru_reward_kernel(const float* __restrict__ obs,
                                  const float* __restrict__ act,
                                  const float* __restrict__ Wih,
                                  const float* __restrict__ bih,
                                  const float* __restrict__ Whh,
                                  const float* __restrict__ bhh,
                                  const float* __restrict__ Wr,
                                  const float* __restrict__ br,
                                  float* __restrict__ outH,
                                  float* __restrict__ outR)
{
    __shared__ float sX[2][16][XS];   // x_t tile, double buffered
    __shared__ float sH[2][16][HS];   // hidden state, double buffered
    __shared__ float sS[16][SS];      // gi+gh (+both biases) for r,z cols [0,128)
    __shared__ float sGI[16][NS];     // gi_n + b_ih_n
    __shared__ float sGH[16][NS];     // gh_n + b_hh_n

    const int tid  = threadIdx.x;
    const int wave = tid >> 5;
    const int lane = tid & 31;
    const int l16  = lane & 15;
    const int koff = (lane >> 4) << 1;        // 0 (lanes 0-15) or 2 (lanes 16-31)
    const int r0   = (lane >> 4) << 3;        // D-tile row base: 0 or 8
    const int b0   = blockIdx.x * 16;         // batch tile base

    // Gate column this wave/lane owns: waves 0-7 cover cols 0..127 (r,z),
    // waves 8-11 cover cols 128..191 (n).
    const int col = (wave < 8 ? wave * 16 : 128 + (wave - 8) * 16) + l16;

    // ---- Preload B-matrix fragments (weights) into VGPRs ----
    // B frag (16x16x4 f32): lanes 0-15 hold (K=k0..k0+1, N=lane),
    // lanes 16-31 hold (K=k0+2..k0+3, N=lane-16).  B[k][n] = W[col n][k].
    v2f bI[9];
    #pragma unroll
    for (int kk = 0; kk < 9; ++kk) {
        int k = kk * 4 + koff;
        bI[kk].x = Wih[col * DIN + k];
        bI[kk].y = Wih[col * DIN + k + 1];
    }
    v2f bHf[16];
    #pragma unroll
    for (int kk = 0; kk < 16; ++kk) {
        int k = kk * 4 + koff;
        bHf[kk].x = Whh[col * HH + k];
        bHf[kk].y = Whh[col * HH + k + 1];
    }
    const float biasI = bih[col];
    const float biasH = bhh[col];
    const float biasS = biasI + biasH;

    // Reward weights: each lane covers 4 h-columns.
    const float wr0 = Wr[l16 * 4 + 0];
    const float wr1 = Wr[l16 * 4 + 1];
    const float wr2 = Wr[l16 * 4 + 2];
    const float wr3 = Wr[l16 * 4 + 3];
    const float rbias = br[0];

    // x-tile element mapping: e in [0,576): [0,512)=obs (m=e>>5,d=e&31),
    // [512,576)=action (m=(e-512)>>2, d=32+((e-512)&3)).
    auto fetchX = [&](int e, int t) -> float {
        if (e < 512) {
            int m = e >> 5, d = e & 31;
            return obs[((size_t)(b0 + m) * TT + t) * DOBS + d];
        } else {
            int q = e - 512; int m = q >> 2, d = q & 3;
            return act[((size_t)(b0 + m) * TT + t) * DACT + d];
        }
    };
    auto storeX = [&](int buf, int e, float v) {
        if (e < 512) { sX[buf][e >> 5][e & 31] = v; }
        else         { int q = e - 512; sX[buf][q >> 2][DOBS + (q & 3)] = v; }
    };

    // ---- init: h = 0, x[0] -> LDS ----
    {
        float* h0 = &sH[0][0][0];
        for (int e = tid; e < 16 * HS; e += 384) h0[e] = 0.0f;
        for (int e = tid; e < 576; e += 384) storeX(0, e, fetchX(e, 0));
    }
    __syncthreads();

    int hc = 0;
    for (int t = 0; t < TT; ++t) {
        const int xc = t & 1;

        // Issue next x tile's global loads into registers (no wait here;
        // the wait lands after the WMMA phase, hiding HBM latency).
        float f0 = 0.0f, f1 = 0.0f;
        if (t + 1 < TT) {
            f0 = fetchX(tid, t + 1);
            if (tid < 192) f1 = fetchX(tid + 384, t + 1);
        }

        // ---- A-fragment LDS loads, batched up front ----
        v2f ax[9];
        #pragma unroll
        for (int kk = 0; kk < 9; ++kk) {
            int k = kk * 4 + koff;
            ax[kk].x = sX[xc][l16][k];
            ax[kk].y = sX[xc][l16][k + 1];
        }
        v2f ah[16];
        #pragma unroll
        for (int kk = 0; kk < 16; ++kk) {
            int k = kk * 4 + koff;
            ah[kk].x = sH[hc][l16][k];
            ah[kk].y = sH[hc][l16][k + 1];
        }

        // ---- WMMA phase: gi (K=36) and gh (K=64) ----
        v8f acc = {};
        #pragma unroll
        for (int kk = 0; kk < 9; ++kk)
            acc = __builtin_amdgcn_wmma_f32_16x16x4_f32(
                false, ax[kk], false, bI[kk], (short)0, acc, false, false);
        if (wave < 8) {
            // r,z columns: chain gh into the same accumulator -> gi+gh
            #pragma unroll
            for (int kk = 0; kk < 16; ++kk)
                acc = __builtin_amdgcn_wmma_f32_16x16x4_f32(
                    false, ah[kk], false, bHf[kk], (short)0, acc, false, false);
            const int c = wave * 16 + l16;
            #pragma unroll
            for (int v = 0; v < 8; ++v) sS[r0 + v][c] = acc[v] + biasS;
        } else {
            // n columns: keep gi_n and gh_n separate (r gates gh_n only)
            const int c = (wave - 8) * 16 + l16;
            #pragma unroll
            for (int v = 0; v < 8; ++v) sGI[r0 + v][c] = acc[v] + biasI;
            v8f acc2 = {};
            #pragma unroll
            for (int kk = 0; kk < 16; ++kk)
                acc2 = __builtin_amdgcn_wmma_f32_16x16x4_f32(
                    false, ah[kk], false, bHf[kk], (short)0, acc2, false, false);
            #pragma unroll
            for (int v = 0; v < 8; ++v) sGH[r0 + v][c] = acc2[v] + biasH;
        }
        __syncthreads();

        // ---- commit prefetched x[t+1] to the other LDS buffer ----
        // (between barrier 1 and barrier 2, so next step's WMMA reads are safe)
        if (t + 1 < TT) {
            storeX(xc ^ 1, tid, f0);
            if (tid < 192) storeX(xc ^ 1, tid + 384, f1);
        }

        // ---- gate phase: 16 rows x 64 h-cols over 384 threads ----
        const int hn = hc ^ 1;
        for (int e = tid; e < 16 * HH; e += 384) {
            int m = e >> 6, j = e & 63;
            float rg = 1.0f / (1.0f + __expf(-sS[m][j]));
            float zg = 1.0f / (1.0f + __expf(-sS[m][64 + j]));
            float ng = tanhf(sGI[m][j] + rg * sGH[m][j]);
            float h  = (1.0f - zg) * ng + zg * sH[hc][m][j];
            sH[hn][m][j] = h;
            if (t == TT - 1) outH[(size_t)(b0 + m) * HH + j] = h;
        }
        __syncthreads();

        // ---- reward phase (waves 0-7; overlaps next step's WMMA) ----
        if (wave < 8) {
            const int row = 2 * wave + (lane >> 4);
            const float* hp = &sH[hn][row][0];
            float p = hp[l16 * 4 + 0] * wr0 + hp[l16 * 4 + 1] * wr1 +
                      hp[l16 * 4 + 2] * wr2 + hp[l16 * 4 + 3] * wr3;
            p += __shfl_xor(p, 1, 32);
            p += __shfl_xor(p, 2, 32);
            p += __shfl_xor(p, 4, 32);
            p += __shfl_xor(p, 8, 32);
            if (l16 == 0) outR[(size_t)(b0 + row) * TT + t] = p + rbias;
        }
        hc = hn;
    }
}

extern "C" void kernel_launch(void* const* d_in, const int* in_sizes, int n_in,
                              void* d_out, int out_size, void* d_ws, size_t ws_size,
                              hipStream_t stream) {
    const float* obs = (const float*)d_in[0];
    const float* act = (const float*)d_in[1];
    const float* Wih = (const float*)d_in[2];
    const float* bih = (const float*)d_in[3];
    const float* Whh = (const float*)d_in[4];
    const float* bhh = (const float*)d_in[5];
    const float* Wr  = (const float*)d_in[6];
    const float* br  = (const float*)d_in[7];
    float* outH = (float*)d_out;                       // [B, H]
    float* outR = outH + (size_t)BATCH * HH;           // [B, T, 1]
    (void)in_sizes; (void)n_in; (void)out_size; (void)d_ws; (void)ws_size;
    dim3 grid(BATCH / 16);
    dim3 block(384);
    hipLaunchKernelGGL(gru_reward_kernel, grid, block, 0, stream,
                       obs, act, Wih, bih, Whh, bhh, Wr, br, outH, outR);
}